// NeuMF_34402688041328
// MI455X (gfx1250) — compile-verified
//
#include <hip/hip_runtime.h>

typedef float v2f __attribute__((ext_vector_type(2)));
typedef float v8f __attribute__((ext_vector_type(8)));

// ---- LDS layout (float offsets) ----
#define W1F_OFF 0              // 4096 floats: W1 in B-fragment order [n*16+k][lane][2]
#define W2F_OFF 4096           // 2048 floats: W2 in B-fragment order
#define B1_OFF  6144           // 64
#define B2_OFF  6208           // 32
#define WO_OFF  6240           // 48
#define H1_OFF  6288           // per-wave 16x66 dwords (even base -> b64 aligned)
#define H1_STRIDE 66
#define H1_PER_WAVE (16 * H1_STRIDE)      // 1056
#define RD_OFF  (H1_OFF + 8 * H1_PER_WAVE) // 14736: 16 row-dots per wave (128)
#define SMEM_TOTAL (RD_OFF + 128)          // 14864 floats = 59456 B

__global__ __launch_bounds__(256)
void neumf_kernel(const int* __restrict__ user, const int* __restrict__ item,
                  const float* __restrict__ mf_u_tbl, const float* __restrict__ mf_i_tbl,
                  const float* __restrict__ mlp_u_tbl, const float* __restrict__ mlp_i_tbl,
                  const float* __restrict__ W1, const float* __restrict__ b1,
                  const float* __restrict__ W2, const float* __restrict__ b2,
                  const float* __restrict__ Wo, const float* __restrict__ bo,
                  float* __restrict__ out)
{
    __shared__ float smem[SMEM_TOTAL];
    const int tid  = threadIdx.x;
    const int wave = tid >> 5;
    const int lane = tid & 31;
    const int lo   = lane & 15;
    const int hi   = lane >> 4;      // 0 = lanes 0-15, 1 = lanes 16-31

    // ---- Stage weights into LDS in WMMA B-fragment order ----
    // B fragment (4x16 f32): VGPR0 holds K-rows {4k, 4k+2}, VGPR1 {4k+1, 4k+3};
    // element e of lane ln at step k, tile n = W[4k + 2*(ln>=16) + e][16n + (ln&15)]
    for (int idx = tid; idx < 4096; idx += 256) {
        int e = idx & 1, ln = (idx >> 1) & 31, kk = (idx >> 6) & 15, n = idx >> 10;
        int c = 16 * n + (ln & 15);
        int r = 4 * kk + 2 * (ln >> 4) + e;
        smem[W1F_OFF + idx] = W1[r * 64 + c];
    }
    for (int idx = tid; idx < 2048; idx += 256) {
        int e = idx & 1, ln = (idx >> 1) & 31, kk = (idx >> 6) & 15, n = idx >> 10;
        int c = 16 * n + (ln & 15);
        int r = 4 * kk + 2 * (ln >> 4) + e;
        smem[W2F_OFF + idx] = W2[r * 32 + c];
    }
    if (tid < 64) smem[B1_OFF + tid] = b1[tid];
    if (tid < 32) smem[B2_OFF + tid] = b2[tid];
    if (tid < 48) smem[WO_OFF + tid] = Wo[tid];
    __syncthreads();

    const int rbase = blockIdx.x * 128 + wave * 16;

    // ---- Gather mlp_in directly from global in WMMA A-fragment layout ----
    // A (16x4 f32): lane holds row M=(lane&15); VGPR0/1 = K = 4k + 2*(lane>=16) + {0,1}
    const int u  = user[rbase + lo];
    const int it = item[rbase + lo];
    const float* urow = mlp_u_tbl + (size_t)u  * 32;
    const float* irow = mlp_i_tbl + (size_t)it * 32;
    v2f a[16];
    #pragma unroll
    for (int k = 0; k < 8; ++k)
        a[k] = *(const v2f*)(urow + 4 * k + 2 * hi);        // cols 0..31  (user half)
    #pragma unroll
    for (int k = 0; k < 8; ++k)
        a[8 + k] = *(const v2f*)(irow + 4 * k + 2 * hi);    // cols 32..63 (item half)

    float* h1 = &smem[H1_OFF + wave * H1_PER_WAVE];
    const v2f* w1f = (const v2f*)&smem[W1F_OFF];
    const v2f* w2f = (const v2f*)&smem[W2F_OFF];

    // ---- Layer 1: h1 = relu(mlp_in @ W1 + b1), 4 N-tiles of 16 ----
    #pragma unroll
    for (int n = 0; n < 4; ++n) {
        float bv = smem[B1_OFF + 16 * n + lo];
        v8f acc = {bv, bv, bv, bv, bv, bv, bv, bv};
        #pragma unroll
        for (int k = 0; k < 16; ++k) {
            v2f b = w1f[(n * 16 + k) * 32 + lane];   // ds_load_b64, conflict-free
            acc = __builtin_amdgcn_wmma_f32_16x16x4_f32(
                false, a[k], false, b, (short)0, acc, false, false);
        }
        #pragma unroll
        for (int v = 0; v < 8; ++v) {
            float x = acc[v] > 0.f ? acc[v] : 0.f;
            h1[(v + 8 * hi) * H1_STRIDE + 16 * n + lo] = x;   // D-layout -> row-major LDS
        }
    }
    asm volatile("" ::: "memory");   // keep ds stores before reloads (HW DS is in-order per wave)

    // ---- Reload h1 as A-fragments for layer 2 ----
    #pragma unroll
    for (int k = 0; k < 16; ++k)
        a[k] = *(const v2f*)(h1 + lo * H1_STRIDE + 4 * k + 2 * hi);

    // ---- Layer 2: h2 = relu(h1 @ W2 + b2), 2 N-tiles ----
    v8f acc2[2];
    #pragma unroll
    for (int n = 0; n < 2; ++n) {
        float bv = smem[B2_OFF + 16 * n + lo];
        v8f acc = {bv, bv, bv, bv, bv, bv, bv, bv};
        #pragma unroll
        for (int k = 0; k < 16; ++k) {
            v2f b = w2f[(n * 16 + k) * 32 + lane];
            acc = __builtin_amdgcn_wmma_f32_16x16x4_f32(
                false, a[k], false, b, (short)0, acc, false, false);
        }
        #pragma unroll
        for (int v = 0; v < 8; ++v) acc[v] = acc[v] > 0.f ? acc[v] : 0.f;
        acc2[n] = acc;
    }

    // ---- Output head: row-dot(h2, Wo[16:48]) via width-16 shuffle reduction ----
    float woA = smem[WO_OFF + 16 + lo];
    float woB = smem[WO_OFF + 32 + lo];
    float p[8];
    #pragma unroll
    for (int v = 0; v < 8; ++v)
        p[v] = acc2[0][v] * woA + acc2[1][v] * woB;
    #pragma unroll
    for (int off = 1; off < 16; off <<= 1) {
        #pragma unroll
        for (int v = 0; v < 8; ++v)
            p[v] += __shfl_xor(p[v], off, 16);
    }
    if (lo == 0) {
        #pragma unroll
        for (int v = 0; v < 8; ++v)
            smem[RD_OFF + wave * 16 + 8 * hi + v] = p[v];   // lane0 -> rows 0-7, lane16 -> rows 8-15
    }
    asm volatile("" ::: "memory");

    // ---- GMF branch + final sum: lanes 0-15 each own one batch row ----
    if (lane < 16) {
        int gr = rbase + lane;
        int uu = user[gr], ii = item[gr];
        const float* mu = mf_u_tbl + (size_t)uu * 16;
        const float* mi = mf_i_tbl + (size_t)ii * 16;
        float s = bo[0] + smem[RD_OFF + wave * 16 + lane];
        #pragma unroll
        for (int c = 0; c < 16; c += 4) {
            float4 x = *(const float4*)(mu + c);
            float4 y = *(const float4*)(mi + c);
            s += x.x * y.x * smem[WO_OFF + c]     + x.y * y.y * smem[WO_OFF + c + 1]
               + x.z * y.z * smem[WO_OFF + c + 2] + x.w * y.w * smem[WO_OFF + c + 3];
        }
        out[gr] = s;
    }
}

extern "C" void kernel_launch(void* const* d_in, const int* in_sizes, int n_in,
                              void* d_out, int out_size, void* d_ws, size_t ws_size,
                              hipStream_t stream) {
    (void)n_in; (void)out_size; (void)d_ws; (void)ws_size;
    const int*   user  = (const int*)d_in[0];
    const int*   item  = (const int*)d_in[1];
    const float* mf_u  = (const float*)d_in[2];
    const float* mf_i  = (const float*)d_in[3];
    const float* mlp_u = (const float*)d_in[4];
    const float* mlp_i = (const float*)d_in[5];
    const float* W1    = (const float*)d_in[6];
    const float* b1    = (const float*)d_in[7];
    const float* W2    = (const float*)d_in[8];
    const float* b2    = (const float*)d_in[9];
    const float* Wo    = (const float*)d_in[10];
    const float* bo    = (const float*)d_in[11];
    const int batch = in_sizes[0];           // 1,048,576
    const int blocks = batch / 128;          // 128 rows per block (8 waves x 16 rows)
    neumf_kernel<<<blocks, 256, 0, stream>>>(user, item, mf_u, mf_i, mlp_u, mlp_i,
                                             W1, b1, W2, b2, Wo, bo, (float*)d_out);
}